// SimpleSSM_32100585570717
// MI455X (gfx1250) — compile-verified
//
#include <hip/hip_runtime.h>

#define BSZ 8
#define TLEN 4096
#define DM 1024
#define LCH 64              // time-chunk length
#define NCHUNK (TLEN / LCH) // 64
#define LN_EPS 1e-5f

typedef float v2f __attribute__((ext_vector_type(2)));
typedef float v8f __attribute__((ext_vector_type(8)));

// ---------------------------------------------------------------------------
// Kernel 1: per-(batch, chunk) local end-state with zero initial state.
// grid (NCHUNK, BSZ) x 1024 threads; thread d owns channel d.
// ---------------------------------------------------------------------------
__global__ __launch_bounds__(1024)
void ssm_chunk_state(const float* __restrict__ x,
                     const float* __restrict__ A,
                     const float* __restrict__ B,
                     float* __restrict__ sloc) {
  const int d = threadIdx.x;
  const int c = blockIdx.x;
  const int b = blockIdx.y;
  const float a  = A[d];
  const float bb = B[d];
  const float* xp = x + ((size_t)b * TLEN + (size_t)c * LCH) * DM + d;
  float s = 0.f;
#pragma unroll 8
  for (int i = 0; i < LCH; ++i) {
    s = fmaf(a, s, bb * xp[(size_t)i * DM]);   // h = A*h + B*x (coalesced 4KB rows)
  }
  sloc[((size_t)b * NCHUNK + c) * DM + d] = s;
}

// ---------------------------------------------------------------------------
// Kernel 2: sequential prefix over the 64 chunks per (b, d).
// h_init[c] = A^64 * h_init[c-1] + sloc[c-1]; h_init[0] = 0.
// ---------------------------------------------------------------------------
__global__ __launch_bounds__(256)
void ssm_chunk_scan(const float* __restrict__ A,
                    const float* __restrict__ sloc,
                    float* __restrict__ hinit) {
  const int tid = blockIdx.x * blockDim.x + threadIdx.x;   // 0 .. 8191
  const int b = tid / DM;
  const int d = tid % DM;
  float aL = A[d];
  aL *= aL; aL *= aL; aL *= aL; aL *= aL; aL *= aL; aL *= aL;  // A^64
  const float* sp = sloc  + (size_t)b * NCHUNK * DM + d;
  float*       hp = hinit + (size_t)b * NCHUNK * DM + d;
  float h = 0.f;
  for (int c = 0; c < NCHUNK; ++c) {
    hp[(size_t)c * DM] = h;
    h = fmaf(aL, h, sp[(size_t)c * DM]);
  }
}

// ---------------------------------------------------------------------------
// Kernel 3: recompute chunk with correct h_init, y-tile in LDS, fused LN.
// LN row reductions on matrix cores: V_WMMA_F32_16X16X4_F32 ones-trick.
// grid (NCHUNK, BSZ) x 1024 threads; dynamic LDS = 64*1024 floats + partials.
// ---------------------------------------------------------------------------
__global__ __launch_bounds__(1024)
void ssm_y_ln(const float* __restrict__ x,
              const float* __restrict__ A,  const float* __restrict__ Bv,
              const float* __restrict__ Cv, const float* __restrict__ Dv,
              const float* __restrict__ gamma, const float* __restrict__ beta,
              const float* __restrict__ hinit,
              float* __restrict__ out) {
  extern __shared__ float smem[];
  float* yT    = smem;                 // [64][1024] y tile
  float* psum  = smem + LCH * DM;      // [64][8] partial row sums
  float* psq   = psum + LCH * 8;       // [64][8] partial row sums of squares
  float* meanv = psq  + LCH * 8;       // [64]
  float* rstdv = meanv + LCH;          // [64]

  const int d = threadIdx.x;
  const int c = blockIdx.x;
  const int b = blockIdx.y;

  // ---- Phase A: recurrence with correct initial state; stash y in LDS ----
  {
    const float a  = A[d],  bb = Bv[d];
    const float cc = Cv[d], dd = Dv[d];
    float h = hinit[((size_t)b * NCHUNK + c) * DM + d];
    const float* xp = x + ((size_t)b * TLEN + (size_t)c * LCH) * DM + d;
#pragma unroll 4
    for (int i = 0; i < LCH; ++i) {
      const float xv = xp[(size_t)i * DM];
      h = fmaf(a, h, bb * xv);
      yT[i * DM + d] = fmaf(cc, h, dd * xv);
    }
  }
  __syncthreads();

  // ---- Phase B: WMMA row-sum / row-sumsq over channels ----
  // 32 waves: tg = wave/8 picks 16 t-rows, kg = wave%8 picks 128 channels.
  // f32 16x16x4 A layout: lanes 0-15 hold (M=lane, K=0,1); lanes 16-31 hold
  // (M=lane-16, K=2,3) -> one ds_load_b64 of 2 consecutive channels per lane.
  {
    const int wave = d >> 5;
    const int lane = d & 31;
    const int tg   = wave >> 3;
    const int kg   = wave & 7;
    const int m    = lane & 15;
    const int koff = (lane >> 4) * 2;
    const int rowbase = (tg * 16 + m) * DM;

    v8f csum = {0.f, 0.f, 0.f, 0.f, 0.f, 0.f, 0.f, 0.f};
    v8f csq  = {0.f, 0.f, 0.f, 0.f, 0.f, 0.f, 0.f, 0.f};
    v2f ones = {1.f, 1.f};

    for (int kt = 0; kt < 32; ++kt) {
      const int k0 = kg * 128 + kt * 4 + koff;
      v2f av = *(const v2f*)&yT[rowbase + k0];
      v2f a2 = av * av;
      csum = __builtin_amdgcn_wmma_f32_16x16x4_f32(
          false, av, false, ones, (short)0, csum, false, false);
      csq  = __builtin_amdgcn_wmma_f32_16x16x4_f32(
          false, a2, false, ones, (short)0, csq,  false, false);
    }
    // C layout: VGPR v, lanes 0-15 -> row v (col = lane); lanes 16-31 -> row v+8.
    // Every column holds the same row-sum, so lane 0 / lane 16 can each dump 8 rows.
    if (lane == 0) {
#pragma unroll
      for (int v = 0; v < 8; ++v) {
        psum[(tg * 16 + v) * 8 + kg] = csum[v];
        psq [(tg * 16 + v) * 8 + kg] = csq[v];
      }
    } else if (lane == 16) {
#pragma unroll
      for (int v = 0; v < 8; ++v) {
        psum[(tg * 16 + 8 + v) * 8 + kg] = csum[v];
        psq [(tg * 16 + 8 + v) * 8 + kg] = csq[v];
      }
    }
  }
  __syncthreads();

  // ---- Phase C: finalize mean / rstd per t-row ----
  if (d < LCH) {
    float s = 0.f, q = 0.f;
#pragma unroll
    for (int k = 0; k < 8; ++k) { s += psum[d * 8 + k]; q += psq[d * 8 + k]; }
    const float mu  = s * (1.f / DM);
    const float var = q * (1.f / DM) - mu * mu;
    meanv[d] = mu;
    rstdv[d] = rsqrtf(var + LN_EPS);
  }
  __syncthreads();

  // ---- Phase D: normalize + coalesced store ----
  {
    const float g  = gamma[d];
    const float be = beta[d];
    float* op = out + ((size_t)b * TLEN + (size_t)c * LCH) * DM + d;
#pragma unroll 4
    for (int i = 0; i < LCH; ++i) {
      const float yv = yT[i * DM + d];
      op[(size_t)i * DM] = fmaf((yv - meanv[i]) * rstdv[i], g, be);
    }
  }
}

// ---------------------------------------------------------------------------
extern "C" void kernel_launch(void* const* d_in, const int* in_sizes, int n_in,
                              void* d_out, int out_size, void* d_ws, size_t ws_size,
                              hipStream_t stream) {
  const float* x     = (const float*)d_in[0];
  const float* A     = (const float*)d_in[1];
  const float* B     = (const float*)d_in[2];
  const float* C     = (const float*)d_in[3];
  const float* D     = (const float*)d_in[4];
  const float* gamma = (const float*)d_in[5];
  const float* beta  = (const float*)d_in[6];
  float* out = (float*)d_out;

  float* sloc  = (float*)d_ws;                        // [BSZ][NCHUNK][DM] (2 MiB)
  float* hinit = sloc + (size_t)BSZ * NCHUNK * DM;    // [BSZ][NCHUNK][DM] (2 MiB)

  dim3 grid(NCHUNK, BSZ);

  ssm_chunk_state<<<grid, DM, 0, stream>>>(x, A, B, sloc);
  ssm_chunk_scan<<<(BSZ * DM) / 256, 256, 0, stream>>>(A, sloc, hinit);

  const size_t smem_bytes =
      (size_t)(LCH * DM + LCH * 8 * 2 + LCH * 2) * sizeof(float);  // ~260 KB < 320 KB/WGP
  ssm_y_ln<<<grid, DM, smem_bytes, stream>>>(x, A, B, C, D, gamma, beta,
                                             hinit, out);
}